// GradientLayer_45896020525436
// MI455X (gfx1250) — compile-verified
//
#include <hip/hip_runtime.h>
#include <hip/hip_bf16.h>

typedef float v2f __attribute__((ext_vector_type(2)));
typedef float v8f __attribute__((ext_vector_type(8)));

// f32 WMMA: D(16x16) = A(16x4) * B(4x16) + C, full f32 precision.
__device__ __forceinline__ v8f wmma_f32(v2f a, v2f b, v8f c) {
  return __builtin_amdgcn_wmma_f32_16x16x4_f32(false, a, false, b, (short)0, c,
                                               false, false);
}

#define NSTREAM 6   // 0:value 1:d/dx 2:d2/dx2 3:d/dy 4:d2/dy2 5:d/dt
#define H 128

// One hidden layer (128->128) for all 6 dual streams of a 16-sample tile.
// actin/actout are distinct LDS buffers (ping-pong), so no intra-layer barrier.
// B fragments for an N-tile are hoisted into 64 VGPRs and reused by all streams.
__device__ __forceinline__ void mlp_hidden_layer(
    const float* __restrict__ W, const float* __restrict__ bb,
    const float (&actin)[NSTREAM * 16 * H], float (&actout)[NSTREAM * 16 * H],
    const int ln, const int half, const int koff)
{
  #pragma unroll
  for (int j = 0; j < 8; ++j) {                 // N-tiles of 16
    // B 4x16 frag: vgpr r, lane-half h -> row K = 2h + r, col N = 16j + ln
    v2f bf[32];
    #pragma unroll
    for (int kc = 0; kc < 32; ++kc) {
      const int kb = 4 * kc + koff;
      bf[kc].x = W[(size_t)kb * H + 16 * j + ln];
      bf[kc].y = W[(size_t)(kb + 1) * H + 16 * j + ln];
    }
    const float bj = bb[16 * j + ln];           // bias feeds value stream only

    v8f acur, zcur;                             // tanh(a), raw z_d for this tile
    #pragma unroll
    for (int s = 0; s < NSTREAM; ++s) {
      v8f acc;
      #pragma unroll
      for (int v = 0; v < 8; ++v) acc[v] = (s == 0) ? bj : 0.f;

      #pragma unroll
      for (int kc = 0; kc < 32; ++kc) {
        // A 16x4 frag: lanes 0-15 (M=ln, K=0,1), lanes 16-31 (K=2,3)
        const v2f af = *(const v2f*)&actin[(s * 16 + ln) * H + 4 * kc + koff];
        acc = wmma_f32(af, bf[kc], acc);
      }

      // elementwise dual-number rules (D layout: vgpr v -> row v + 8*half)
      v8f res;
      if (s == 0) {
        #pragma unroll
        for (int v = 0; v < 8; ++v) res[v] = tanhf(acc[v]);
        acur = res;
      } else if (s == 1 || s == 3) {            // first-order spatial: keep raw z_d
        zcur = acc;
        #pragma unroll
        for (int v = 0; v < 8; ++v) {
          const float a = acur[v];
          res[v] = (1.f - a * a) * acc[v];
        }
      } else if (s == 2 || s == 4) {            // diagonal second order
        #pragma unroll
        for (int v = 0; v < 8; ++v) {
          const float a  = acur[v];
          const float s2 = 1.f - a * a;
          const float zd = zcur[v];
          res[v] = s2 * acc[v] - 2.f * a * s2 * zd * zd;
        }
      } else {                                  // time tangent
        #pragma unroll
        for (int v = 0; v < 8; ++v) {
          const float a = acur[v];
          res[v] = (1.f - a * a) * acc[v];
        }
      }
      #pragma unroll
      for (int v = 0; v < 8; ++v)
        actout[(s * 16 + (v + 8 * half)) * H + 16 * j + ln] = res[v];
    }
  }
}

__global__ __launch_bounds__(32) void pinn_grad_kernel(
    const float* __restrict__ x,
    const float* __restrict__ W0, const float* __restrict__ b0,
    const float* __restrict__ W1, const float* __restrict__ b1,
    const float* __restrict__ W2, const float* __restrict__ b2,
    const float* __restrict__ W3, const float* __restrict__ b3,
    float* __restrict__ out, int B)
{
  __shared__ __align__(16) float actA[NSTREAM * 16 * H];  // 48 KB
  __shared__ __align__(16) float actB[NSTREAM * 16 * H];  // 48 KB
  __shared__ float z3buf[NSTREAM * 16 * 4];

  const int lane = threadIdx.x & 31;
  const int half = lane >> 4;   // 0 | 1
  const int ln   = lane & 15;
  const int koff = half * 2;    // K-pair select for A and B fragments
  const int base = blockIdx.x * 16;

  // ---------------- layer 0 : 3 -> 128, seed dual streams ----------------
  for (int idx = lane; idx < 16 * H; idx += 32) {
    const int m = idx >> 7, n = idx & (H - 1);
    const float* xs = x + (size_t)(base + m) * 3;
    const float x0 = xs[0], x1 = xs[1], x2 = xs[2];
    const float w0 = W0[0 * H + n], w1 = W0[1 * H + n], w2 = W0[2 * H + n];
    const float z  = fmaf(x0, w0, fmaf(x1, w1, fmaf(x2, w2, b0[n])));
    const float a  = tanhf(z);
    const float s2 = 1.f - a * a;
    actA[(0 * 16 + m) * H + n] = a;                       // value
    actA[(1 * 16 + m) * H + n] = s2 * w0;                 // a_x
    actA[(2 * 16 + m) * H + n] = -2.f * a * s2 * w0 * w0; // a_xx (z_xx = 0)
    actA[(3 * 16 + m) * H + n] = s2 * w1;                 // a_y
    actA[(4 * 16 + m) * H + n] = -2.f * a * s2 * w1 * w1; // a_yy
    actA[(5 * 16 + m) * H + n] = s2 * w2;                 // a_t (TDIM = 2)
  }
  __syncthreads();

  // ---------------- layers 1,2 : 128 -> 128 on the WMMA pipe -------------
  mlp_hidden_layer(W1, b1, actA, actB, ln, half, koff);
  __syncthreads();
  mlp_hidden_layer(W2, b2, actB, actA, ln, half, koff);
  __syncthreads();

  // ---------------- layer 3 : 128 -> 4 (linear) ---------------------------
  {
    const int sg = half;                      // lane half handles 3 streams each
    const int m  = ln;
    #pragma unroll
    for (int ss = 0; ss < 3; ++ss) {
      const int s = 3 * sg + ss;
      float o0 = 0.f, o1 = 0.f, o2 = 0.f, o3 = 0.f;
      #pragma unroll 4
      for (int k = 0; k < H; ++k) {
        const float h = actA[(s * 16 + m) * H + k];
        o0 = fmaf(h, W3[k * 4 + 0], o0);
        o1 = fmaf(h, W3[k * 4 + 1], o1);
        o2 = fmaf(h, W3[k * 4 + 2], o2);
        o3 = fmaf(h, W3[k * 4 + 3], o3);
      }
      if (s == 0) { o0 += b3[0]; o1 += b3[1]; o2 += b3[2]; o3 += b3[3]; }
      z3buf[(s * 16 + m) * 4 + 0] = o0;
      z3buf[(s * 16 + m) * 4 + 1] = o1;
      z3buf[(s * 16 + m) * 4 + 2] = o2;
      z3buf[(s * 16 + m) * 4 + 3] = o3;
    }
  }
  __syncthreads();

  // ---------------- assemble the 17 reference outputs ---------------------
  if (lane < 16) {
    const size_t i = (size_t)base + lane;
    float V[NSTREAM][4];
    #pragma unroll
    for (int s = 0; s < NSTREAM; ++s)
      #pragma unroll
      for (int o = 0; o < 4; ++o)
        V[s][o] = z3buf[(s * 16 + lane) * 4 + o];

    const float c   = V[0][0], v1 = V[0][1], v2 = V[0][2], Fi = V[0][3];
    const float dx0 = V[1][0], dx1 = V[1][1], dx2 = V[1][2], dx3 = V[1][3];
    const float xx0 = V[2][0], xx1 = V[2][1], xx2 = V[2][2], xx3 = V[2][3];
    const float dy0 = V[3][0], dy1 = V[3][1], dy2 = V[3][2], dy3 = V[3][3];
    const float yy0 = V[4][0], yy1 = V[4][1], yy2 = V[4][2], yy3 = V[4][3];
    const float dt0 = V[5][0], dt1 = V[5][1];

    const size_t Bz = (size_t)B;
    out[0 * Bz + i]          = c;                        // c
    out[1 * Bz + i]          = dt0;                      // c_t
    out[2 * Bz + 2 * i + 0]  = dx0;                      // c_grd
    out[2 * Bz + 2 * i + 1]  = dy0;
    out[4 * Bz + 2 * i + 0]  = v1;                       // v
    out[4 * Bz + 2 * i + 1]  = v2;
    out[6 * Bz + i]          = dt1;                      // v_t
    out[7 * Bz + i]          = dx1 + dy2;                // v_div
    out[8 * Bz + 2 * i + 0]  = dx1 * v1 + dy1 * v2;      // v_adv = v_grd @ v
    out[8 * Bz + 2 * i + 1]  = dx2 * v1 + dy2 * v2;
    out[10 * Bz + 2 * i + 0] = xx1 + yy1;                // v_lap
    out[10 * Bz + 2 * i + 1] = xx2 + yy2;
    out[12 * Bz + i]         = Fi;                       // Fi
    out[13 * Bz + 2 * i + 0] = dx3;                      // Fi_grd
    out[13 * Bz + 2 * i + 1] = dy3;
    out[15 * Bz + i]         = xx3 + yy3;                // Fi_lap
    // j = -D grad(c) - c grad(Fi) + c v ;  div j:
    out[16 * Bz + i] = -0.5f * (xx0 + yy0)
                       - (dx0 * dx3 + dy0 * dy3) - c * (xx3 + yy3)
                       + (dx0 * v1 + dy0 * v2) + c * (dx1 + dy2);
  }
}

extern "C" void kernel_launch(void* const* d_in, const int* in_sizes, int n_in,
                              void* d_out, int out_size, void* d_ws, size_t ws_size,
                              hipStream_t stream) {
  const float* x  = (const float*)d_in[0];
  const float* W0 = (const float*)d_in[1];
  const float* b0 = (const float*)d_in[2];
  const float* W1 = (const float*)d_in[3];
  const float* b1 = (const float*)d_in[4];
  const float* W2 = (const float*)d_in[5];
  const float* b2 = (const float*)d_in[6];
  const float* W3 = (const float*)d_in[7];
  const float* b3 = (const float*)d_in[8];

  const int B = in_sizes[0] / 3;       // 65536
  dim3 grid(B / 16), block(32);        // one wave32 per 16-sample tile
  pinn_grad_kernel<<<grid, block, 0, stream>>>(x, W0, b0, W1, b1, W2, b2, W3, b3,
                                               (float*)d_out, B);
}